// OuterProductMean_30365418782823
// MI455X (gfx1250) — compile-verified
//
#include <hip/hip_runtime.h>

#define S_DIM 128
#define N_DIM 384
#define CM    64
#define CC    32
#define CZ    128
#define LN_EPS 1e-5f

// LDS row strides chosen for bank-conflict avoidance (64 banks):
//  sA  stride 40  -> lanes 0-15 (K=s0,s0+1) and 16-31 (K=s0+2,s0+3) hit disjoint bank sets
//  sAB stride 1028 (== 4 mod 64) -> stage-2 A reads: lane groups land on {4m} vs {4m+2} banks
#define SA_STRIDE  40
#define SAB_STRIDE 1028

typedef __attribute__((ext_vector_type(2))) float v2f;
typedef __attribute__((ext_vector_type(8))) float v8f;

__device__ __forceinline__ v8f wmma_f32(v2f a, v2f b, v8f c) {
  // D = A(16x4, f32) * B(4x16, f32) + C(16x16, f32)
  return __builtin_amdgcn_wmma_f32_16x16x4_f32(false, a, false, b, (short)0, c,
                                               false, false);
}

__device__ __forceinline__ float wave_sum(float v) {
#pragma unroll
  for (int off = 16; off > 0; off >>= 1) v += __shfl_xor(v, off, 32);
  return v;
}

// ---------------------------------------------------------------------------
// Stage 0: LayerNorm + projections. One wave (32 lanes) per (s,n) row.
// m_a[row,c] = (LN(msa[row,:]) @ w_a)[c] * mask[row]   (c == lane, CC == 32)
// ---------------------------------------------------------------------------
__global__ __launch_bounds__(256) void opm_ln_proj(
    const float* __restrict__ msa, const float* __restrict__ mask,
    const float* __restrict__ gamma, const float* __restrict__ beta,
    const float* __restrict__ wa, const float* __restrict__ wb,
    float* __restrict__ mA, float* __restrict__ mB) {
  __shared__ float sY[8 * CM];
  const int t = threadIdx.x, wave = t >> 5, lane = t & 31;
  const int row = blockIdx.x * 8 + wave;  // row in [0, S*N)

  const float* x = msa + (size_t)row * CM;
  const float x0 = x[lane], x1 = x[lane + 32];
  const float mean = wave_sum(x0 + x1) * (1.0f / CM);
  const float d0 = x0 - mean, d1 = x1 - mean;
  const float var = wave_sum(d0 * d0 + d1 * d1) * (1.0f / CM);
  const float inv = rsqrtf(var + LN_EPS);
  sY[wave * CM + lane]      = d0 * inv * gamma[lane] + beta[lane];
  sY[wave * CM + lane + 32] = d1 * inv * gamma[lane + 32] + beta[lane + 32];
  __syncthreads();

  const float mk = mask[row];
  float a = 0.f, b = 0.f;
#pragma unroll 8
  for (int m = 0; m < CM; ++m) {
    const float ym = sY[wave * CM + m];  // LDS broadcast
    a = fmaf(ym, wa[m * CC + lane], a);
    b = fmaf(ym, wb[m * CC + lane], b);
  }
  mA[(size_t)row * CC + lane] = a * mk;
  mB[(size_t)row * CC + lane] = b * mk;
}

// ---------------------------------------------------------------------------
// Main fused kernel: block = (i, j-tile of 16), 256 threads = 8 waves.
// Stage 1: ab_j[32,32] = sum_s m_a[s,i,:]^T m_b[s,j,:]  (WMMA f32 16x16x4)
// Stage 2: z[j,0:128]  = ab_flat[16,1024] @ w_out       (WMMA f32 16x16x4)
// Epilogue: (z + b_out) / (norm_ij + 1e-3)
// ---------------------------------------------------------------------------
__global__ __launch_bounds__(256) void opm_main(
    const float* __restrict__ mA, const float* __restrict__ mB,
    const float* __restrict__ mask, const float* __restrict__ wOut,
    const float* __restrict__ bOut, float* __restrict__ out) {
  extern __shared__ float smem[];
  float* sA    = smem;                              // S_DIM * SA_STRIDE
  float* sAB   = sA + S_DIM * SA_STRIDE;            // 16 * SAB_STRIDE
  float* pnorm = sAB + 16 * SAB_STRIDE;             // 256
  float* snorm = pnorm + 256;                       // 16

  const int t = threadIdx.x;
  const int wave = t >> 5, lane = t & 31;
  const int i = blockIdx.y;
  const int j0 = blockIdx.x * 16;

  // Stage m_a[:, i, :] into LDS as [s][c] with padded stride.
  for (int idx = t; idx < S_DIM * CC; idx += 256) {
    const int s = idx >> 5, c = idx & 31;
    sA[s * SA_STRIDE + c] = mA[((size_t)s * N_DIM + i) * CC + c];
  }
  // norm[i, j0+jj] partials: thread t -> jj = t&15, s-chunk = t>>4 (8 s each)
  {
    const int jj = t & 15, chunk = t >> 4;
    float p = 0.f;
#pragma unroll
    for (int s = chunk * 8; s < chunk * 8 + 8; ++s)
      p += mask[s * N_DIM + i] * mask[s * N_DIM + j0 + jj];
    pnorm[t] = p;
  }
  __syncthreads();
  if (t < 16) {
    float sum = 0.f;
#pragma unroll
    for (int c = 0; c < 16; ++c) sum += pnorm[c * 16 + t];
    snorm[t] = sum;
  }

  // ---- Stage 1: each wave computes ab for 2 j's -------------------------
  // WMMA 32-bit A 16x4 layout: VGPR0 lanes0-15 K=s0, lanes16-31 K=s0+2; VGPR1: +1
  const int dBase = lane & 15;
  const int sOff  = (lane >> 4) << 1;  // 0 | 2
  const int mAdd  = (lane >> 4) << 3;  // D-tile: lanes16-31 hold M = r+8
  for (int jj = wave * 2; jj < wave * 2 + 2; ++jj) {
    const int j = j0 + jj;
    v8f acc00 = {}, acc01 = {}, acc10 = {}, acc11 = {};
    for (int s0 = 0; s0 < S_DIM; s0 += 4) {
      const int sa = s0 + sOff;
      v2f a0, a1, b0, b1;
      a0.x = sA[sa * SA_STRIDE + dBase];            // A[c, s] = m_a[s, i, c]
      a0.y = sA[(sa + 1) * SA_STRIDE + dBase];
      a1.x = sA[sa * SA_STRIDE + 16 + dBase];
      a1.y = sA[(sa + 1) * SA_STRIDE + 16 + dBase];
      const float* bp = mB + ((size_t)sa * N_DIM + j) * CC;  // B[s, d]
      b0.x = bp[dBase];
      b0.y = bp[(size_t)N_DIM * CC + dBase];
      b1.x = bp[16 + dBase];
      b1.y = bp[(size_t)N_DIM * CC + 16 + dBase];
      acc00 = wmma_f32(a0, b0, acc00);
      acc01 = wmma_f32(a0, b1, acc01);
      acc10 = wmma_f32(a1, b0, acc10);
      acc11 = wmma_f32(a1, b1, acc11);
    }
    float* abj = sAB + jj * SAB_STRIDE;  // ab flattened as k = c*32 + d
#pragma unroll
    for (int r = 0; r < 8; ++r) {
      const int c = r + mAdd;
      abj[c * CC + dBase]             = acc00[r];
      abj[c * CC + 16 + dBase]        = acc01[r];
      abj[(c + 16) * CC + dBase]      = acc10[r];
      abj[(c + 16) * CC + 16 + dBase] = acc11[r];
    }
  }
  __syncthreads();

  // ---- Stage 2: Z[16j, z0:z0+16] = sAB[16,1024] @ w_out[:, z0:z0+16] ----
  const int z0 = wave * 16;
  const int kOff = (lane >> 4) << 1;  // 0 | 2
  const int jr = lane & 15;
  v8f accA = {}, accB = {};
  for (int k0 = 0; k0 < CC * CC; k0 += 8) {  // two interleaved WMMA chains
    v2f a, b;
    a.x = sAB[jr * SAB_STRIDE + k0 + kOff];
    a.y = sAB[jr * SAB_STRIDE + k0 + kOff + 1];
    const float* wp = wOut + (size_t)(k0 + kOff) * CZ + z0 + dBase;
    b.x = wp[0];
    b.y = wp[CZ];
    accA = wmma_f32(a, b, accA);

    a.x = sAB[jr * SAB_STRIDE + k0 + 4 + kOff];
    a.y = sAB[jr * SAB_STRIDE + k0 + 4 + kOff + 1];
    const float* wp2 = wOut + (size_t)(k0 + 4 + kOff) * CZ + z0 + dBase;
    b.x = wp2[0];
    b.y = wp2[CZ];
    accB = wmma_f32(a, b, accB);
  }
  const v8f accZ = accA + accB;

  const float bo = bOut[z0 + dBase];
#pragma unroll
  for (int r = 0; r < 8; ++r) {
    const int jj = r + mAdd;  // D-tile M index
    const float nrm = snorm[jj] + 0.001f;
    out[((size_t)i * N_DIM + (j0 + jj)) * CZ + z0 + dBase] = (accZ[r] + bo) / nrm;
  }
}

// ---------------------------------------------------------------------------
extern "C" void kernel_launch(void* const* d_in, const int* in_sizes, int n_in,
                              void* d_out, int out_size, void* d_ws,
                              size_t ws_size, hipStream_t stream) {
  (void)in_sizes; (void)n_in; (void)out_size; (void)ws_size;
  const float* msa   = (const float*)d_in[0];
  const float* mask  = (const float*)d_in[1];
  const float* gamma = (const float*)d_in[2];
  const float* beta  = (const float*)d_in[3];
  const float* wa    = (const float*)d_in[4];
  const float* wb    = (const float*)d_in[5];
  const float* wout  = (const float*)d_in[6];
  const float* bout  = (const float*)d_in[7];
  float* out = (float*)d_out;

  float* mAbuf = (float*)d_ws;                        // [S,N,32] = 6.3 MB
  float* mBbuf = mAbuf + (size_t)S_DIM * N_DIM * CC;  // [S,N,32] = 6.3 MB

  opm_ln_proj<<<(S_DIM * N_DIM) / 8, 256, 0, stream>>>(msa, mask, gamma, beta,
                                                       wa, wb, mAbuf, mBbuf);

  const size_t ldsBytes =
      (size_t)(S_DIM * SA_STRIDE + 16 * SAB_STRIDE + 256 + 16) * sizeof(float);
  opm_main<<<dim3(N_DIM / 16, N_DIM), 256, ldsBytes, stream>>>(
      mAbuf, mBbuf, mask, wout, bout, out);
}